// GCNEncoder_26852135534760
// MI455X (gfx1250) — compile-verified
//
#include <hip/hip_runtime.h>
#include <stdint.h>

typedef __attribute__((ext_vector_type(2))) float v2f;
typedef __attribute__((ext_vector_type(8))) float v8f;

#define CH_IN  128
#define CH_HID 128
#define CH_OUT 64

// ---------------------------------------------------------------- init / util

__global__ void init_deg_kernel(float* __restrict__ deg, int n) {
    int i = blockIdx.x * blockDim.x + threadIdx.x;
    if (i < n) deg[i] = 1.0f;                   // self-loop contributes 1 to in-degree
}

__global__ void zero_kernel(float* __restrict__ p, long long n) {
    long long i = (long long)blockIdx.x * blockDim.x + threadIdx.x;
    if (i < n) p[i] = 0.0f;
}

__global__ void deg_accum_kernel(const long long* __restrict__ dstI,
                                 float* __restrict__ deg, int E) {
    int e = blockIdx.x * blockDim.x + threadIdx.x;
    if (e < E) {
        int d = (int)dstI[e];
        unsafeAtomicAdd(&deg[d], 1.0f);         // native GLOBAL_ATOMIC_ADD_F32
    }
}

__global__ void rsqrt_kernel(float* __restrict__ deg, int n) {
    int i = blockIdx.x * blockDim.x + threadIdx.x;
    if (i < n) deg[i] = rsqrtf(deg[i]);         // deg >= 1 always (self-loop)
}

// ---------------------------------------------------------------- WMMA GEMM
// Y[M,N] = X[M,K] @ W[K,N], fp32 via V_WMMA_F32_16X16X4_F32.
// One wave computes one 16x16 tile; K-loop steps by 4.
// A-frag (16x4): lane&15 = row, VGPR v holds K = k + 2*(lane>>4) + v.
// B-frag (4x16): lane&15 = col, VGPR v holds K = k + 2*(lane>>4) + v.
// D (16x16 f32): VGPR r, lane -> M = r + 8*(lane>>4), N = lane&15.
template <int N, int K>
__global__ __launch_bounds__(256) void gemm_wmma_f32(const float* __restrict__ X,
                                                     const float* __restrict__ W,
                                                     float* __restrict__ Y, int M) {
    const int wavesPerBlock = blockDim.x >> 5;
    const int tileId = blockIdx.x * wavesPerBlock + (threadIdx.x >> 5);
    const int tilesN = N / 16;
    const int tilesM = M / 16;                  // M is a multiple of 16 (50000 = 3125*16)
    if (tileId >= tilesM * tilesN) return;      // wave-uniform exit: EXEC all-ones inside

    const int mt   = tileId / tilesN;
    const int nt   = tileId % tilesN;
    const int lane = threadIdx.x & 31;
    const int half = lane >> 4;                 // 0: lanes 0-15, 1: lanes 16-31
    const int l16  = lane & 15;
    const int row  = mt * 16 + l16;
    const int col  = nt * 16 + l16;

    v8f acc = {};
    for (int k = 0; k < K; k += 4) {
        const int ka = k + 2 * half;            // even -> 8B aligned
        v2f a = *reinterpret_cast<const v2f*>(X + (long long)row * K + ka);
        v2f b;
        b.x = W[(long long)ka * N + col];
        b.y = W[(long long)(ka + 1) * N + col];
        acc = __builtin_amdgcn_wmma_f32_16x16x4_f32(
            /*neg_a=*/false, a, /*neg_b=*/false, b,
            /*c_mod=*/(short)0, acc, /*reuse_a=*/false, /*reuse_b=*/false);
    }

#pragma unroll
    for (int r = 0; r < 8; ++r) {
        Y[(long long)(mt * 16 + r + 8 * half) * N + col] = acc[r];
    }
}

// ---------------------------------------------------------------- edge scatter
// agg[dst] += t[src] * dinv[src]*dinv[dst].
// Exactly ONE wave per edge; each lane handles C/32 floats (float4 for C=128,
// float2 for C=64). src/dst are wave-uniform -> forced into SGPRs via
// readfirstlane so dinv lookups and address bases become scalar ops; the
// gather and the atomics stay fully contiguous per wave.
template <int C>
__global__ __launch_bounds__(256) void scatter_kernel(const float* __restrict__ t,
                                                      const long long* __restrict__ srcI,
                                                      const long long* __restrict__ dstI,
                                                      const float* __restrict__ dinv,
                                                      float* __restrict__ agg, int E) {
    constexpr int VEC = C / 32;
    typedef float VecT __attribute__((ext_vector_type(VEC)));

    long long gid  = (long long)blockIdx.x * blockDim.x + threadIdx.x;
    long long e    = gid >> 5;                  // wave id == edge id
    if (e >= E) return;
    const int lane = threadIdx.x & 31;
    const int c    = lane * VEC;

    int s = (int)srcI[e];
    int d = (int)dstI[e];
    s = __builtin_amdgcn_readfirstlane(s);      // wave-uniform -> SGPR
    d = __builtin_amdgcn_readfirstlane(d);
    const float nrm = dinv[s] * dinv[d];        // scalar loads / scalar mul

    const VecT v = *reinterpret_cast<const VecT*>(t + (long long)s * C + c);
    float* o = agg + (long long)d * C + c;
#pragma unroll
    for (int j = 0; j < VEC; ++j) {
        unsafeAtomicAdd(o + j, v[j] * nrm);     // native GLOBAL_ATOMIC_ADD_F32
    }
}

// ---------------------------------------------------------------- finalize
// agg[i,c] = act(agg[i,c] + t[i,c]*dinv[i]^2 + bias[c])   (self-loop + bias)
template <int C, bool RELU>
__global__ __launch_bounds__(256) void finalize_kernel(float* __restrict__ agg,
                                                       const float* __restrict__ t,
                                                       const float* __restrict__ dinv,
                                                       const float* __restrict__ bias,
                                                       int Nn) {
    long long gid = (long long)blockIdx.x * blockDim.x + threadIdx.x;
    if (gid >= (long long)Nn * C) return;
    int node = (int)(gid / C);
    int c    = (int)(gid % C);
    float di = dinv[node];
    float v  = agg[gid] + t[gid] * di * di + bias[c];
    agg[gid] = RELU ? fmaxf(v, 0.0f) : v;
}

// ---------------------------------------------------------------- launch

extern "C" void kernel_launch(void* const* d_in, const int* in_sizes, int n_in,
                              void* d_out, int out_size, void* d_ws, size_t ws_size,
                              hipStream_t stream) {
    const float*     x    = (const float*)d_in[0];
    const long long* ei   = (const long long*)d_in[1];   // int64 edge_index [2, E]
    const float*     W1   = (const float*)d_in[2];
    const float*     b1   = (const float*)d_in[3];
    const float*     W2   = (const float*)d_in[4];
    const float*     b2   = (const float*)d_in[5];
    float*           out  = (float*)d_out;

    const int Nn = in_sizes[0] / CH_IN;   // 50000
    const int E  = in_sizes[1] / 2;       // 800000
    const long long* srcI = ei;
    const long long* dstI = ei + E;

    // workspace layout (floats)
    float* dinv = (float*)d_ws;                          // Nn          (deg -> dinv in place)
    float* t1   = dinv + Nn;                             // Nn * 128
    float* agg1 = t1 + (size_t)Nn * CH_HID;              // Nn * 128    (agg -> h1 in place)
    float* t2   = agg1 + (size_t)Nn * CH_HID;            // Nn * 64

    const int B = 256;

    // degree / norm
    init_deg_kernel<<<(Nn + B - 1) / B, B, 0, stream>>>(dinv, Nn);
    deg_accum_kernel<<<(E + B - 1) / B, B, 0, stream>>>(dstI, dinv, E);
    rsqrt_kernel<<<(Nn + B - 1) / B, B, 0, stream>>>(dinv, Nn);

    // zero aggregation buffers
    {
        long long n1 = (long long)Nn * CH_HID;
        long long n2 = (long long)Nn * CH_OUT;
        zero_kernel<<<(int)((n1 + B - 1) / B), B, 0, stream>>>(agg1, n1);
        zero_kernel<<<(int)((n2 + B - 1) / B), B, 0, stream>>>(out, n2);
    }

    // layer 1: t1 = x @ W1
    {
        int tiles = (Nn / 16) * (CH_HID / 16);
        int wavesPerBlock = B / 32;
        gemm_wmma_f32<CH_HID, CH_IN><<<(tiles + wavesPerBlock - 1) / wavesPerBlock, B, 0, stream>>>(
            x, W1, t1, Nn);
    }
    // scatter edges (one wave per edge), then self-loop + bias + relu
    {
        long long threads = (long long)E * 32;
        scatter_kernel<CH_HID><<<(int)((threads + B - 1) / B), B, 0, stream>>>(
            t1, srcI, dstI, dinv, agg1, E);
        long long n = (long long)Nn * CH_HID;
        finalize_kernel<CH_HID, true><<<(int)((n + B - 1) / B), B, 0, stream>>>(
            agg1, t1, dinv, b1, Nn);
    }

    // layer 2: t2 = h1 @ W2   (h1 lives in agg1)
    {
        int tiles = (Nn / 16) * (CH_OUT / 16);
        int wavesPerBlock = B / 32;
        gemm_wmma_f32<CH_OUT, CH_HID><<<(tiles + wavesPerBlock - 1) / wavesPerBlock, B, 0, stream>>>(
            agg1, W2, t2, Nn);
    }
    {
        long long threads = (long long)E * 32;
        scatter_kernel<CH_OUT><<<(int)((threads + B - 1) / B), B, 0, stream>>>(
            t2, srcI, dstI, dinv, out, E);
        long long n = (long long)Nn * CH_OUT;
        finalize_kernel<CH_OUT, false><<<(int)((n + B - 1) / B), B, 0, stream>>>(
            out, t2, dinv, b2, Nn);
    }
}